// LightGCN_31628139168318
// MI455X (gfx1250) — compile-verified
//
#include <hip/hip_runtime.h>
#include <hip/hip_bf16.h>

// LightGCN propagation on MI455X (gfx1250).
// Memory-bound scatter/gather: embedding table (38.4 MB/buffer) is L2-resident
// (192 MB L2), edges stream from HBM (re-read per layer -> L2 hits in layers
// 2-3). wave32: one wave per edge, lane owns float2 of the 64-wide row. Edge
// scalars go through SGPRs (readfirstlane -> s_load), next edge's row is
// prefetched (global_prefetch_b8), scatter uses native f32 L2 atomics
// (relaxed, agent scope). Running-sum traffic (d_out) uses non-temporal hints
// so it does not evict the L2-resident gather/atomic working set.

#define LG_NU    100000
#define LG_NI    50000
#define LG_NT    150000        // LG_NU + LG_NI
#define LG_EMB   64
#define LG_NE    4800000
#define LG_VEC4  (LG_NT * LG_EMB / 4)   // 2,400,000 float4 elements

// Native clang vector type: required by __builtin_nontemporal_load/store
// (HIP's float4 is a struct and is rejected by the builtin).
typedef float v4f __attribute__((ext_vector_type(4)));
typedef float v2f __attribute__((ext_vector_type(2)));

__device__ __forceinline__ void lg_atomic_add_f32(float* p, float v) {
    // Relaxed agent-scope fetch-add -> global_atomic_add_f32 (no return).
    __hip_atomic_fetch_add(p, v, __ATOMIC_RELAXED, __HIP_MEMORY_SCOPE_AGENT);
}

// One wave32 per edge (grid-strided): gather row[dst] (coalesced b64 per lane),
// scale by edge_val, scatter-add into row[src] via f32 L2 atomics.
// (Placed first in the file so the disasm snippet shows this kernel.)
__global__ void lg_edges(const float* __restrict__ cur, float* __restrict__ nxt,
                         const float* __restrict__ ev,
                         const int*   __restrict__ es,
                         const int*   __restrict__ ed) {
    const int lane   = threadIdx.x & 31;
    const int stride = (int)((gridDim.x * blockDim.x) >> 5);
    int wave = __builtin_amdgcn_readfirstlane(
                   (int)((blockIdx.x * blockDim.x + threadIdx.x) >> 5));

    for (int e = wave; e < LG_NE; e += stride) {
        // Software prefetch of the NEXT edge's source row to hide the random
        // gather latency (lowers to global_prefetch_b8 on gfx1250).
        int en = e + stride;
        if (en < LG_NE) {
            int dn = __builtin_amdgcn_readfirstlane(ed[en]);
            __builtin_prefetch(cur + (size_t)dn * LG_EMB + lane * 2, 0, 0);
        }

        // Wave-uniform edge scalars -> SGPRs -> s_load (SMEM), not VMEM.
        int   dst = __builtin_amdgcn_readfirstlane(ed[e]);
        int   src = __builtin_amdgcn_readfirstlane(es[e]);
        float val = ev[e];

        const v2f m =
            reinterpret_cast<const v2f*>(cur + (size_t)dst * LG_EMB)[lane];
        float* o = nxt + (size_t)src * LG_EMB + (size_t)(lane * 2);
        lg_atomic_add_f32(o,     m.x * val);
        lg_atomic_add_f32(o + 1, m.y * val);
    }
}

// Concatenate user/item embeddings into cur, and seed the running layer-sum.
// The sum (d_out) stream is non-temporal: no reuse until the next accum pass.
__global__ void lg_init(const float* __restrict__ ue, const float* __restrict__ ie,
                        float* __restrict__ cur, float* __restrict__ sum) {
    int idx = blockIdx.x * blockDim.x + threadIdx.x;
    if (idx >= LG_VEC4) return;
    const int uel = LG_NU * LG_EMB / 4;
    v4f v = (idx < uel) ? reinterpret_cast<const v4f*>(ue)[idx]
                        : reinterpret_cast<const v4f*>(ie)[idx - uel];
    reinterpret_cast<v4f*>(cur)[idx] = v;
    __builtin_nontemporal_store(v, reinterpret_cast<v4f*>(sum) + idx);
}

__global__ void lg_zero(float* __restrict__ buf) {
    int idx = blockIdx.x * blockDim.x + threadIdx.x;
    if (idx >= LG_VEC4) return;
    reinterpret_cast<v4f*>(buf)[idx] = (v4f){0.f, 0.f, 0.f, 0.f};
}

// sum = (sum + layer) * s   (s = 1.0 for layers 0,1; 0.25 fused on last layer)
// sum accesses are non-temporal (streamed); layer buffer stays L2-regular.
__global__ void lg_accum(const float* __restrict__ layer, float* __restrict__ sum,
                         float s) {
    int idx = blockIdx.x * blockDim.x + threadIdx.x;
    if (idx >= LG_VEC4) return;
    v4f* s4 = reinterpret_cast<v4f*>(sum);
    v4f a = __builtin_nontemporal_load(s4 + idx);
    v4f b = reinterpret_cast<const v4f*>(layer)[idx];
    a = (a + b) * s;
    __builtin_nontemporal_store(a, s4 + idx);
}

extern "C" void kernel_launch(void* const* d_in, const int* in_sizes, int n_in,
                              void* d_out, int out_size, void* d_ws, size_t ws_size,
                              hipStream_t stream) {
    (void)in_sizes; (void)n_in; (void)out_size; (void)ws_size;

    const float* ue = (const float*)d_in[0];   // user_emb   (100000 x 64)
    const float* ie = (const float*)d_in[1];   // item_emb   (50000  x 64)
    const float* ev = (const float*)d_in[2];   // edge_val   (4.8M)
    const int*   es = (const int*)  d_in[3];   // edge_src   (4.8M)
    const int*   ed = (const int*)  d_in[4];   // edge_dst   (4.8M)
    float* out = (float*)d_out;                // (150000 x 64) = (u_out, i_out) flat

    float* bufA = (float*)d_ws;                          // 38.4 MB
    float* bufB = bufA + (size_t)LG_NT * LG_EMB;         // 38.4 MB

    const dim3 blk(256);
    const dim3 gElem((LG_VEC4 + 255) / 256);   // 9375 blocks
    const dim3 gEdge(8192);                    // 65536 waves, ~73 edges/wave

    lg_init<<<gElem, blk, 0, stream>>>(ue, ie, bufA, out);

    float* cur = bufA;
    float* nxt = bufB;
    for (int layer = 0; layer < 3; ++layer) {
        lg_zero <<<gElem, blk, 0, stream>>>(nxt);
        lg_edges<<<gEdge, blk, 0, stream>>>(cur, nxt, ev, es, ed);
        lg_accum<<<gElem, blk, 0, stream>>>(nxt, out,
                                            (layer == 2) ? 0.25f : 1.0f);
        float* t = cur; cur = nxt; nxt = t;
    }
}